// PermInvarEventEncoder_19232863552123
// MI455X (gfx1250) — compile-verified
//
#include <hip/hip_runtime.h>
#include <hip/hip_bf16.h>

#define NUM_NODES 10000
#define DIM 64
#define BSZ 32
#define KX  129          // 2*DIM + 1
#define FOURD 256        // 4*DIM
#define XPITCH 132       // padded LDS row pitch (floats)

typedef float v2f __attribute__((ext_vector_type(2)));
typedef float v8f __attribute__((ext_vector_type(8)));

// ---------------------------------------------------------------------------
// Kernel 1: zero a buffer (float4 granularity, grid-stride)
// ---------------------------------------------------------------------------
__global__ __launch_bounds__(256) void zero_kern(float4* __restrict__ p, size_t n4) {
    size_t stride = (size_t)gridDim.x * blockDim.x;
    for (size_t i = (size_t)blockIdx.x * blockDim.x + threadIdx.x; i < n4; i += stride) {
        p[i] = make_float4(0.f, 0.f, 0.f, 0.f);
    }
}

// ---------------------------------------------------------------------------
// Kernel 2: per-event encoder GEMM (f32 WMMA) + scatter-add
//   block = 256 threads = 8 waves, 32 events/block
//   wave w: event-tile (w>>2) of 16 events, output chunk (w&3) of 64 columns
//   chunk 0 -> uu -> out_u   [bs][N][D]   (transposed final layout)
//   chunk 1 -> iu -> out_i   [bs][N][D]
//   chunk 2 -> ug -> glob_u  [bs][D][N]   (node-contiguous for reduction)
//   chunk 3 -> ig -> glob_i  [bs][D][N]
// ---------------------------------------------------------------------------
__global__ __launch_bounds__(256) void encode_scatter(
    const int*   __restrict__ users,
    const int*   __restrict__ items,
    const float* __restrict__ messages,
    const float* __restrict__ user_emb,
    const float* __restrict__ item_emb,
    const float* __restrict__ W,      // [129][256] row-major
    const float* __restrict__ bvec,   // [256]
    float* __restrict__ out_u,
    float* __restrict__ out_i,
    float* __restrict__ glob_u,
    float* __restrict__ glob_i)
{
    __shared__ float xs[32][XPITCH];  // 32 event rows of x = [ue|ie|msg]
    __shared__ int   s_user[32];
    __shared__ int   s_item[32];

    const int tid   = threadIdx.x;
    const int ebase = blockIdx.x * 32;

    // ---- stage x rows + scatter indices into LDS (8 threads per event) ----
    {
        const int el = tid >> 3;          // event slot 0..31
        const int l8 = tid & 7;           // 0..7
        const int e  = ebase + el;
        const int su = users[e];
        const int si = items[e];
        if (l8 == 0) {
            s_user[el] = su;
            s_item[el] = si;
            xs[el][2 * DIM] = messages[e];
        }
        const float4* up = (const float4*)(user_emb + (size_t)(su % NUM_NODES) * DIM);
        const float4* ip = (const float4*)(item_emb + (size_t)(si % NUM_NODES) * DIM);
        float4 a0 = up[l8 * 2], a1 = up[l8 * 2 + 1];
        float4 c0 = ip[l8 * 2], c1 = ip[l8 * 2 + 1];
        *(float4*)&xs[el][l8 * 8]           = a0;   // row pitch 132 floats = 16B multiple
        *(float4*)&xs[el][l8 * 8 + 4]       = a1;
        *(float4*)&xs[el][DIM + l8 * 8]     = c0;
        *(float4*)&xs[el][DIM + l8 * 8 + 4] = c1;
    }
    __syncthreads();

    // ---- wave-level WMMA GEMM: 16 events x 64 cols, K = 129 ----
    const int wave  = tid >> 5;
    const int lane  = tid & 31;
    const int tileE = wave >> 2;          // 0..1
    const int chunk = wave & 3;           // 0..3  (uu/iu/ug/ig)
    const int m16   = lane & 15;
    const int hi    = lane >> 4;          // 0 or 1 (K-pair select per ISA layout)
    const int cbase = chunk * DIM;        // column base in [0,256)

    v8f acc[4];
    #pragma unroll
    for (int nt = 0; nt < 4; ++nt) {
        float bv = bvec[cbase + nt * 16 + m16];   // bias depends on column only
        #pragma unroll
        for (int r = 0; r < 8; ++r) acc[nt][r] = bv;
    }

    const float* xrow = &xs[tileE * 16 + m16][0];

    // K = 0..127 via 32 steps of V_WMMA_F32_16X16X4_F32 per N-tile
    #pragma unroll 4
    for (int kb = 0; kb < 2 * DIM; kb += 4) {
        const int k = kb + hi * 2;
        v2f a;
        a.x = xrow[k];                    // A: lane row = lane%16, K pair by lane half
        a.y = xrow[k + 1];
        const float* Wr = W + (size_t)k * FOURD + cbase + m16;
        #pragma unroll
        for (int nt = 0; nt < 4; ++nt) {
            v2f bf;
            bf.x = Wr[nt * 16];           // B row k,   col = nt*16 + lane%16
            bf.y = Wr[nt * 16 + FOURD];   // B row k+1
            acc[nt] = __builtin_amdgcn_wmma_f32_16x16x4_f32(
                false, a, false, bf, (short)0, acc[nt], false, false);
        }
    }

    // ---- k = 128 fixup: enc += msg * W[128][col]  (scalar FMA) ----
    #pragma unroll
    for (int nt = 0; nt < 4; ++nt) {
        const float w128 = W[(size_t)(2 * DIM) * FOURD + cbase + nt * 16 + m16];
        #pragma unroll
        for (int r = 0; r < 8; ++r) {
            // acc row M = r + 8*hi  (C/D layout), its message column:
            float msg = xs[tileE * 16 + r + 8 * hi][2 * DIM];
            acc[nt][r] += msg * w128;
        }
    }

    // ---- scatter-add (wave-uniform destination select) ----
    float* dst;
    const int* sArr;
    if (chunk == 0)      { dst = out_u;  sArr = s_user; }
    else if (chunk == 1) { dst = out_i;  sArr = s_item; }
    else if (chunk == 2) { dst = glob_u; sArr = s_user; }
    else                 { dst = glob_i; sArr = s_item; }
    const bool to_out = (chunk < 2);

    #pragma unroll
    for (int r = 0; r < 8; ++r) {
        const int m  = r + 8 * hi;                // event row in tile
        const int s  = sArr[tileE * 16 + m];      // flat state index in [0, N*bs)
        const int n  = s >> 5;                    // s / bs
        const int bb = s & 31;                    // s % bs
        #pragma unroll
        for (int nt = 0; nt < 4; ++nt) {
            const int col = nt * 16 + m16;        // 0..63 within chunk
            size_t off;
            if (to_out) off = ((size_t)bb * NUM_NODES + n) * DIM + col;      // [bs][N][D]
            else        off = ((size_t)bb * DIM + col) * NUM_NODES + n;      // [bs][D][N]
            atomicAdd(dst + off, acc[nt][r]);
        }
    }
}

// ---------------------------------------------------------------------------
// Kernel 3: gmax[p] = max_n glob[p][n]  for p in [0, 2*bs*D), contiguous N run
// ---------------------------------------------------------------------------
__global__ __launch_bounds__(256) void max_reduce(const float* __restrict__ glob,
                                                  float* __restrict__ gmax)
{
    const float* src = glob + (size_t)blockIdx.x * NUM_NODES;
    float m = -__builtin_inff();
    for (int i = threadIdx.x; i < NUM_NODES; i += 256) m = fmaxf(m, src[i]);
    __shared__ float red[256];
    red[threadIdx.x] = m;
    __syncthreads();
    #pragma unroll
    for (int s = 128; s > 0; s >>= 1) {
        if (threadIdx.x < s) red[threadIdx.x] = fmaxf(red[threadIdx.x], red[threadIdx.x + s]);
        __syncthreads();
    }
    if (threadIdx.x == 0) gmax[blockIdx.x] = red[0];
}

// ---------------------------------------------------------------------------
// Kernel 4: out[tensor][b][n][d] += gmax[tensor][b][d]
//   blockIdx.y = tensor*32 + b  (64 slices), x grid-strides over N*D
// ---------------------------------------------------------------------------
__global__ __launch_bounds__(256) void bcast_add(float* __restrict__ out,
                                                 const float* __restrict__ gmax)
{
    const int tb = blockIdx.y;                               // 0..63
    float* dst = out + (size_t)tb * (NUM_NODES * DIM);
    __shared__ float g[DIM];
    if (threadIdx.x < DIM)
        g[threadIdx.x] = gmax[(size_t)tb * DIM + threadIdx.x]; // [2][32][64] flat == tb*64+d
    __syncthreads();
    const int total  = NUM_NODES * DIM;
    const int stride = gridDim.x * blockDim.x;
    for (int i = blockIdx.x * blockDim.x + threadIdx.x; i < total; i += stride) {
        dst[i] += g[i & (DIM - 1)];
    }
}

// ---------------------------------------------------------------------------
extern "C" void kernel_launch(void* const* d_in, const int* in_sizes, int n_in,
                              void* d_out, int out_size, void* d_ws, size_t ws_size,
                              hipStream_t stream) {
    (void)n_in; (void)out_size; (void)ws_size;
    const int*   users    = (const int*)d_in[0];
    const int*   items    = (const int*)d_in[1];
    const float* messages = (const float*)d_in[2];
    const float* user_emb = (const float*)d_in[3];
    const float* item_emb = (const float*)d_in[4];
    const float* W        = (const float*)d_in[5];
    const float* bvec     = (const float*)d_in[6];
    const int    E        = in_sizes[0];

    const size_t TEN = (size_t)BSZ * NUM_NODES * DIM;        // 20,480,000 floats per tensor

    float* out   = (float*)d_out;
    float* out_u = out;
    float* out_i = out + TEN;

    float* ws     = (float*)d_ws;
    float* glob_u = ws;                    // [bs][D][N]
    float* glob_i = ws + TEN;              // [bs][D][N]
    float* gmax   = ws + 2 * TEN;          // [2][bs][D] = 4096 floats

    // 1) zero accumulators (d_out and global workspace) every call
    zero_kern<<<2048, 256, 0, stream>>>((float4*)out, (2 * TEN) / 4);
    zero_kern<<<2048, 256, 0, stream>>>((float4*)glob_u, (2 * TEN) / 4);

    // 2) encoder GEMM (f32 WMMA) + scatter-add (E = 500000, divisible by 32)
    encode_scatter<<<E / 32, 256, 0, stream>>>(
        users, items, messages, user_emb, item_emb, W, bvec,
        out_u, out_i, glob_u, glob_i);

    // 3) max over nodes for both global tensors (2*bs*D = 4096 reductions)
    max_reduce<<<2 * BSZ * DIM, 256, 0, stream>>>(glob_u, gmax);

    // 4) broadcast-add pooled max onto states
    dim3 g4(128, 2 * BSZ);
    bcast_add<<<g4, 256, 0, stream>>>(out, gmax);
}